// ResidualGINLayer_13537736917857
// MI455X (gfx1250) — compile-verified
//
#include <hip/hip_runtime.h>
#include <hip/hip_bf16.h>

#define DD 128          // feature dim
#define TE 32           // edges / nodes per block tile (2 x 16-row WMMA subtiles)

typedef float v2f __attribute__((ext_vector_type(2)));
typedef float v8f __attribute__((ext_vector_type(8)));

// D = A(16x4, f32) * B(4x16, f32) + C(16x16, f32)   -- V_WMMA_F32_16X16X4_F32
__device__ __forceinline__ v8f wmma_k4(v2f a, v2f b, v8f c) {
    return __builtin_amdgcn_wmma_f32_16x16x4_f32(
        /*neg_a=*/false, a, /*neg_b=*/false, b,
        /*c_mod=*/(short)0, c, /*reuse_a=*/false, /*reuse_b=*/false);
}

// ---------------------------------------------------------------------------
// Kernel 0: zero workspace (agg + BN accumulators)
// ---------------------------------------------------------------------------
__global__ void zero_ws(float* __restrict__ p, long n) {
    long i = (long)blockIdx.x * blockDim.x + threadIdx.x;
    long stride = (long)gridDim.x * blockDim.x;
    for (; i < n; i += stride) p[i] = 0.0f;
}

// ---------------------------------------------------------------------------
// Kernel 1: edge MLP  h = relu(LN(concat(x[row], edge_attr) @ W1 + b1))
//           then scatter-add into agg[col]
// Block: 256 threads (8 waves), 32 edges per block, 2 M-subtiles per wave
// sharing each B fragment (halves W1 L2 traffic, 1 global load / 2 WMMAs).
// ---------------------------------------------------------------------------
__global__ __launch_bounds__(256) void edge_mlp_scatter(
    const float* __restrict__ x, const float* __restrict__ ea,
    const float* __restrict__ W1, const float* __restrict__ b1,
    const float* __restrict__ g1, const float* __restrict__ be1,
    const int* __restrict__ eidx, float* __restrict__ agg,
    int E_, int N_)
{
    __shared__ float As[TE][2 * DD + 1];   // xc tile, stride 257 (bank-friendly)
    __shared__ float Hs[TE][DD + 1];       // GEMM result tile, stride 129
    __shared__ float red[TE][8];
    __shared__ float red2[TE][8];
    __shared__ float muS[TE], rsS[TE];
    __shared__ int   rowS[TE], colS[TE];

    const int tid = threadIdx.x;
    const int e0  = blockIdx.x * TE;

    if (tid < TE) {
        int e = e0 + tid;
        int ec = e < E_ ? e : (E_ - 1);
        rowS[tid] = eidx[ec];          // edge_index[0][e]
        colS[tid] = eidx[E_ + ec];     // edge_index[1][e]
    }

    // Prefetch next block's edge_attr tile (global_prefetch_b8, speculative)
    if (e0 + 2 * TE <= E_) {
        const char* nxt = (const char*)(ea + (long)(e0 + TE) * DD);
        __builtin_prefetch(nxt + tid * 64, 0, 0);
    }
    __syncthreads();

    // Stage xc = [x[row[e]], edge_attr[e]]  : 32 x 256
    for (int j = 0; j < 32; ++j) {
        int idx = tid + j * 256;           // 0..8191
        int e   = idx >> 8;
        int c   = idx & 255;
        int eg  = e0 + e; if (eg >= E_) eg = E_ - 1;
        float v;
        if (c < DD) v = x[(long)rowS[e] * DD + c];
        else        v = ea[(long)eg * DD + (c - DD)];
        As[e][c] = v;
    }
    __syncthreads();

    // Each wave computes output columns [16*wave, 16*wave+16) for 32 rows
    const int wave = tid >> 5;
    const int lane = tid & 31;
    const int half = lane >> 4;   // 0: K pair {0,1}/rows M0-7 ; 1: K pair {2,3}/rows M8-15
    const int l16  = lane & 15;
    const int n0   = wave * 16;

    v8f acc0 = {}, acc1 = {};
    for (int k0 = 0; k0 < 2 * DD; k0 += 4) {
        v2f b;
        const float* wp = W1 + (long)(k0 + 2 * half) * DD + (n0 + l16);
        b.x = wp[0];
        b.y = wp[DD];
        v2f a0, a1;
        a0.x = As[l16     ][k0 + 2 * half];
        a0.y = As[l16     ][k0 + 2 * half + 1];
        a1.x = As[16 + l16][k0 + 2 * half];
        a1.y = As[16 + l16][k0 + 2 * half + 1];
        acc0 = wmma_k4(a0, b, acc0);      // rows  0..15
        acc1 = wmma_k4(a1, b, acc1);      // rows 16..31
    }

    float bias = b1[n0 + l16];
#pragma unroll
    for (int i = 0; i < 8; ++i) {
        Hs[     i + 8 * half][n0 + l16] = acc0[i] + bias;
        Hs[16 + i + 8 * half][n0 + l16] = acc1[i] + bias;
    }
    __syncthreads();

    // Row-wise LayerNorm over 128 features (8 threads per row)
    {
        int r = tid >> 3, c8 = tid & 7;
        float s = 0.f, q = 0.f;
#pragma unroll
        for (int j = 0; j < 16; ++j) {
            float v = Hs[r][c8 + 8 * j];
            s += v; q += v * v;
        }
        red[r][c8] = s; red2[r][c8] = q;
    }
    __syncthreads();
    if (tid < TE) {
        float s = 0.f, q = 0.f;
        for (int j = 0; j < 8; ++j) { s += red[tid][j]; q += red2[tid][j]; }
        float mu  = s * (1.0f / DD);
        float var = q * (1.0f / DD) - mu * mu;
        muS[tid] = mu;
        rsS[tid] = rsqrtf(var + 1e-5f);
    }
    __syncthreads();

    // normalize + relu + scatter-add (global_atomic_add_f32)
    {
        int r = tid >> 3, c8 = tid & 7;
        if ((e0 + r) < E_) {
            long dstBase = (long)colS[r] * DD;
            float mu = muS[r], rs = rsS[r];
#pragma unroll
            for (int j = 0; j < 16; ++j) {
                int d = c8 + 8 * j;
                float v = (Hs[r][d] - mu) * rs * g1[d] + be1[d];
                v = fmaxf(v, 0.0f);
                atomicAdd(&agg[dstBase + d], v);
            }
        }
    }
}

// ---------------------------------------------------------------------------
// Kernel 2: node MLP
//   pre = (1+eps)*x + agg
//   u   = relu(LN(pre @ W2 + b2))
//   out = u @ W3 + b3 + 2*x          (pre-BN), plus BN partial sums
// Block: 256 threads (8 waves), 32 nodes per block, 2 subtiles per wave.
// ---------------------------------------------------------------------------
__global__ __launch_bounds__(256) void node_mlp(
    const float* __restrict__ x, const float* __restrict__ agg,
    const float* __restrict__ W2, const float* __restrict__ b2,
    const float* __restrict__ g2, const float* __restrict__ be2,
    const float* __restrict__ W3, const float* __restrict__ b3,
    const float* __restrict__ eps, float* __restrict__ out,
    float* __restrict__ bnsum, float* __restrict__ bnsq, int N_)
{
    __shared__ float P[TE][DD + 1];
    __shared__ float U[TE][DD + 1];
    __shared__ float red[TE][8];
    __shared__ float red2[TE][8];
    __shared__ float muS[TE], rsS[TE];

    const int tid = threadIdx.x;
    const long nb = (long)blockIdx.x * TE;
    const float opE = 1.0f + eps[0];

    // Stage pre = (1+eps)*x + agg : 32 x 128
    for (int j = 0; j < 16; ++j) {
        int idx = tid + j * 256;        // 0..4095
        int m = idx >> 7, d = idx & 127;
        long node = nb + m; if (node >= N_) node = N_ - 1;
        long g = node * DD + d;
        P[m][d] = opE * x[g] + agg[g];
    }
    __syncthreads();

    const int wave = tid >> 5;
    const int lane = tid & 31;
    const int half = lane >> 4;
    const int l16  = lane & 15;
    const int n0   = wave * 16;

    // GEMM 1: pre @ W2 (K = 128)
    v8f acc0 = {}, acc1 = {};
    for (int k0 = 0; k0 < DD; k0 += 4) {
        v2f b;
        const float* wp = W2 + (long)(k0 + 2 * half) * DD + (n0 + l16);
        b.x = wp[0];
        b.y = wp[DD];
        v2f a0, a1;
        a0.x = P[l16     ][k0 + 2 * half];
        a0.y = P[l16     ][k0 + 2 * half + 1];
        a1.x = P[16 + l16][k0 + 2 * half];
        a1.y = P[16 + l16][k0 + 2 * half + 1];
        acc0 = wmma_k4(a0, b, acc0);
        acc1 = wmma_k4(a1, b, acc1);
    }
    float bias = b2[n0 + l16];
#pragma unroll
    for (int i = 0; i < 8; ++i) {
        U[     i + 8 * half][n0 + l16] = acc0[i] + bias;
        U[16 + i + 8 * half][n0 + l16] = acc1[i] + bias;
    }
    __syncthreads();

    // LayerNorm + ReLU on U (in place), 8 threads per row
    {
        int r = tid >> 3, c8 = tid & 7;
        float s = 0.f, q = 0.f;
#pragma unroll
        for (int j = 0; j < 16; ++j) {
            float v = U[r][c8 + 8 * j];
            s += v; q += v * v;
        }
        red[r][c8] = s; red2[r][c8] = q;
    }
    __syncthreads();
    if (tid < TE) {
        float s = 0.f, q = 0.f;
        for (int j = 0; j < 8; ++j) { s += red[tid][j]; q += red2[tid][j]; }
        float mu  = s * (1.0f / DD);
        float var = q * (1.0f / DD) - mu * mu;
        muS[tid] = mu;
        rsS[tid] = rsqrtf(var + 1e-5f);
    }
    __syncthreads();
    {
        int r = tid >> 3, c8 = tid & 7;
        float mu = muS[r], rs = rsS[r];
#pragma unroll
        for (int j = 0; j < 16; ++j) {
            int d = c8 + 8 * j;
            float v = (U[r][d] - mu) * rs * g2[d] + be2[d];
            U[r][d] = fmaxf(v, 0.0f);
        }
    }
    __syncthreads();

    // GEMM 2: u @ W3 (K = 128), + b3 + 2*x, write out + BN partial sums
    v8f acc2 = {}, acc3 = {};
    for (int k0 = 0; k0 < DD; k0 += 4) {
        v2f b;
        const float* wp = W3 + (long)(k0 + 2 * half) * DD + (n0 + l16);
        b.x = wp[0];
        b.y = wp[DD];
        v2f a0, a1;
        a0.x = U[l16     ][k0 + 2 * half];
        a0.y = U[l16     ][k0 + 2 * half + 1];
        a1.x = U[16 + l16][k0 + 2 * half];
        a1.y = U[16 + l16][k0 + 2 * half + 1];
        acc2 = wmma_k4(a0, b, acc2);
        acc3 = wmma_k4(a1, b, acc3);
    }
    float b3v = b3[n0 + l16];
    float s = 0.f, q = 0.f;
#pragma unroll
    for (int i = 0; i < 8; ++i) {
        int m0 = i + 8 * half;
        long node0 = nb + m0;
        if (node0 < N_) {
            long g = node0 * DD + (n0 + l16);
            float v = acc2[i] + b3v + 2.0f * x[g];
            out[g] = v;
            s += v; q += v * v;
        }
        long node1 = nb + 16 + m0;
        if (node1 < N_) {
            long g = node1 * DD + (n0 + l16);
            float v = acc3[i] + b3v + 2.0f * x[g];
            out[g] = v;
            s += v; q += v * v;
        }
    }
    atomicAdd(&bnsum[n0 + l16], s);
    atomicAdd(&bnsq [n0 + l16], q);
}

// ---------------------------------------------------------------------------
// Kernel 3: BatchNorm apply (batch statistics, biased variance)
// ---------------------------------------------------------------------------
__global__ void bn_apply(float* __restrict__ out,
                         const float* __restrict__ bnsum,
                         const float* __restrict__ bnsq,
                         const float* __restrict__ bn_g,
                         const float* __restrict__ bn_b,
                         long total, float invN)
{
    long i = (long)blockIdx.x * blockDim.x + threadIdx.x;
    long stride = (long)gridDim.x * blockDim.x;
    for (; i < total; i += stride) {
        int d = (int)(i & (DD - 1));
        float mu  = bnsum[d] * invN;
        float var = bnsq[d] * invN - mu * mu;
        out[i] = (out[i] - mu) * rsqrtf(var + 1e-5f) * bn_g[d] + bn_b[d];
    }
}

// ---------------------------------------------------------------------------
extern "C" void kernel_launch(void* const* d_in, const int* in_sizes, int n_in,
                              void* d_out, int out_size, void* d_ws, size_t ws_size,
                              hipStream_t stream) {
    const float* x    = (const float*)d_in[0];
    const float* ea   = (const float*)d_in[1];
    const float* W1   = (const float*)d_in[2];
    const float* b1   = (const float*)d_in[3];
    const float* g1   = (const float*)d_in[4];
    const float* be1  = (const float*)d_in[5];
    const float* W2   = (const float*)d_in[6];
    const float* b2   = (const float*)d_in[7];
    const float* g2   = (const float*)d_in[8];
    const float* be2  = (const float*)d_in[9];
    const float* W3   = (const float*)d_in[10];
    const float* b3   = (const float*)d_in[11];
    const float* eps  = (const float*)d_in[12];
    const float* bn_g = (const float*)d_in[13];
    const float* bn_b = (const float*)d_in[14];
    const int*   eidx = (const int*)d_in[15];   // edge_index, [2, E] flat

    const int Nn = in_sizes[0] / DD;
    const int Ee = in_sizes[1] / DD;

    float* ws    = (float*)d_ws;
    float* agg   = ws;                          // N*128 floats
    float* bnsum = ws + (long)Nn * DD;          // 128 floats
    float* bnsq  = bnsum + DD;                  // 128 floats
    float* out   = (float*)d_out;

    long zn = (long)Nn * DD + 2 * DD;
    zero_ws<<<2048, 256, 0, stream>>>(ws, zn);

    edge_mlp_scatter<<<(Ee + TE - 1) / TE, 256, 0, stream>>>(
        x, ea, W1, b1, g1, be1, eidx, agg, Ee, Nn);

    node_mlp<<<(Nn + TE - 1) / TE, 256, 0, stream>>>(
        x, agg, W2, b2, g2, be2, W3, b3, eps, out, bnsum, bnsq, Nn);

    bn_apply<<<2048, 256, 0, stream>>>(
        out, bnsum, bnsq, bn_g, bn_b, (long)Nn * DD, 1.0f / (float)Nn);
}